// BRO_37495064494800
// MI455X (gfx1250) — compile-verified
//
#include <hip/hip_runtime.h>
#include <math.h>

typedef float v2f __attribute__((ext_vector_type(2)));
typedef float v8f __attribute__((ext_vector_type(8)));
typedef unsigned int u32x4 __attribute__((ext_vector_type(4)));
typedef int i32x4 __attribute__((ext_vector_type(4)));
typedef int i32x8 __attribute__((ext_vector_type(8)));

__device__ __forceinline__ v8f wmma4(v2f a, v2f b, v8f c) {
  // D(16x16,f32) = A(16x4,f32) * B(4x16,f32) + C
  return __builtin_amdgcn_wmma_f32_16x16x4_f32(false, a, false, b, (short)0, c, false, false);
}

#if defined(__has_builtin)
#if __has_builtin(__builtin_amdgcn_tensor_load_to_lds) && \
    __has_builtin(__builtin_amdgcn_s_wait_tensorcnt)
#define OCN_TDM 1
#endif
#endif

#ifdef OCN_TDM
// 1D TDM tile load: nelem fp32 words, contiguous in memory -> LDS at lds_byte.
// D# built per CDNA5 ISA 8.3/8.4: group0 = {count, lds_addr, global_addr, type=2},
// group1 = {data_size=4B, tensor_dim0=nelem, tensor_dim1=1, tile_dim0=nelem,
// tensor_dim0_stride=nelem}; groups 2/3 zero (<=2D tensor).
__device__ __forceinline__ void ocn_tdm_load_1d(const float* gsrc, unsigned lds_byte,
                                                int nelem) {
  unsigned long long ga = (unsigned long long)(size_t)gsrc;
  u32x4 g0;
  g0[0] = 1u;                                            // count=1, user mode
  g0[1] = lds_byte;                                      // lds_addr (bytes)
  g0[2] = (unsigned)ga;                                  // global_addr[31:0]
  g0[3] = (unsigned)((ga >> 32) & 0x01ffffffull) | (2u << 30);  // addr[56:32] | type=2
  i32x8 g1;
  g1[0] = 2 << 16;                                       // data_size = 4 bytes
  g1[1] = (nelem & 0xffff) << 16;                        // tensor_dim0[15:0]
  g1[2] = ((nelem >> 16) & 0xffff) | (1 << 16);          // tensor_dim0[31:16], dim1=1
  g1[3] = (nelem & 0xffff) << 16;                        // tile_dim0
  g1[4] = 0;                                             // tile_dim1/2 unused
  g1[5] = nelem;                                         // tensor_dim0_stride[31:0]
  g1[6] = 0;
  g1[7] = 0;
  i32x4 gz = {0, 0, 0, 0};
#if __has_include(<hip/amd_detail/amd_gfx1250_TDM.h>)
  i32x8 gz8 = {0, 0, 0, 0, 0, 0, 0, 0};
  __builtin_amdgcn_tensor_load_to_lds(g0, g1, gz, gz, gz8, 0);
#else
  __builtin_amdgcn_tensor_load_to_lds(g0, g1, gz, gz, 0);
#endif
}
#endif

#define BB   64
#define CC   256
#define HH   32
#define NN   34
#define NFQ  18
#define QQ   612
#define CHUNK 34

// per-frequency scratch sublayout (floats)
#define PQ_VRE 0
#define PQ_VIM 65536
#define PQ_ARE 131072
#define PQ_AIM 196608
#define PQ_TRE 262144
#define PQ_TIM 278528
#define PQ_SRE 294912
#define PQ_SIM 311296
#define PERQ   327680

// dft-matrix region sublayout (floats)
#define M_D1RE 0
#define M_D1IM 576
#define M_F2RE 1152
#define M_F2IM 2240
#define M_G1RE 3328
#define M_G1IM 4416
#define M_C2RE 5504
#define M_C2IM 6080

// workspace layout (float offsets)
#define W_MATS 0
#define W_XFRE 8192UL
#define SZ_Q   10027008UL              // 612*256*64
#define W_XFIM (W_XFRE + SZ_Q)
#define W_ZFRE (W_XFIM + SZ_Q)
#define W_ZFIM (W_ZFRE + SZ_Q)
#define W_YRE  (W_ZFIM + SZ_Q)
#define SZ_Y   9437184UL               // 64*256*32*18
#define W_YIM  (W_YRE + SZ_Y)
#define W_SCR  (W_YIM + SZ_Y)

#define TWO_PI 6.28318530717958647692f

// ---------------------------------------------------------------------------
__global__ void ocn_init_dft(float* mats) {
  const float s = rsqrtf(34.0f);
  for (int i = threadIdx.x; i < HH * NFQ; i += blockDim.x) {   // D1: 32 x 18
    int c = i / NFQ, f = i % NFQ;
    float th = TWO_PI * (float)((c + 1) * f) / 34.0f;
    mats[M_D1RE + i] =  cosf(th) * s;
    mats[M_D1IM + i] = -sinf(th) * s;
  }
  for (int i = threadIdx.x; i < NN * HH; i += blockDim.x) {    // F2: 34 x 32
    int k = i / HH, r = i % HH;
    float th = TWO_PI * (float)((r + 1) * k) / 34.0f;
    mats[M_F2RE + i] =  cosf(th) * s;
    mats[M_F2IM + i] = -sinf(th) * s;
  }
  for (int i = threadIdx.x; i < HH * NN; i += blockDim.x) {    // G1: 32 x 34 (inverse)
    int r = i / NN, k = i % NN;
    float th = TWO_PI * (float)(k * (r + 1)) / 34.0f;
    mats[M_G1RE + i] =  cosf(th) * s;
    mats[M_G1IM + i] =  sinf(th) * s;
  }
  for (int i = threadIdx.x; i < NFQ * HH; i += blockDim.x) {   // C2: 18 x 32 (inv rfft)
    int f = i / HH, c = i % HH;
    float w = (f == 0 || f == NFQ - 1) ? 1.0f : 2.0f;
    float th = TWO_PI * (float)(f * (c + 1)) / 34.0f;
    mats[M_C2RE + i] =  w * cosf(th) * s;
    mats[M_C2IM + i] = -w * sinf(th) * s;
  }
}

// ---------------------------------------------------------------------------
// Forward FFT stage 1 as one giant WMMA GEMM (524288 x 32) * (32 x 36).
__global__ void ocn_fwd1(const float* __restrict__ x, const float* __restrict__ mats,
                         float* __restrict__ yre, float* __restrict__ yim) {
  __shared__ float Bs[HH][48];
  int tid = threadIdx.x;
  for (int i = tid; i < HH * 48; i += 256) {
    int k = i / 48, col = i % 48;
    float v = 0.0f;
    if (col < 18)      v = mats[M_D1RE + k * NFQ + col];
    else if (col < 36) v = mats[M_D1IM + k * NFQ + (col - 18)];
    Bs[k][col] = v;
  }
  __syncthreads();

  int wave = tid >> 5, lane = tid & 31;
  int half = lane >> 4, mrow = lane & 15, koff = half * 2, ncol = lane & 15;
  long row0 = ((long)blockIdx.x * 8 + wave) * 16;

  v8f acc0 = {}, acc1 = {}, acc2 = {};
  const float* xr = x + (size_t)(row0 + mrow) * HH;
  for (int ks = 0; ks < 8; ++ks) {
    int kb = ks * 4 + koff;
    v2f a;  a.x = xr[kb];             a.y = xr[kb + 1];
    v2f b0; b0.x = Bs[kb][ncol];      b0.y = Bs[kb + 1][ncol];
    v2f b1; b1.x = Bs[kb][16 + ncol]; b1.y = Bs[kb + 1][16 + ncol];
    v2f b2; b2.x = Bs[kb][32 + ncol]; b2.y = Bs[kb + 1][32 + ncol];
    acc0 = wmma4(a, b0, acc0);
    acc1 = wmma4(a, b1, acc1);
    acc2 = wmma4(a, b2, acc2);
  }
#pragma unroll
  for (int i = 0; i < 8; ++i) {
    size_t R = (size_t)(row0 + i + half * 8);
    yre[R * NFQ + ncol] = acc0[i];
    int c1 = 16 + ncol;
    if (c1 < 18) yre[R * NFQ + c1] = acc1[i];
    else         yim[R * NFQ + (c1 - 18)] = acc1[i];
    int c2 = 32 + ncol;
    if (c2 < 36) yim[R * NFQ + (c2 - 18)] = acc2[i];
  }
}

// ---------------------------------------------------------------------------
__global__ void ocn_fwd2(const float* __restrict__ yre, const float* __restrict__ yim,
                         const float* __restrict__ mats,
                         float* __restrict__ xfre, float* __restrict__ xfim) {
  const float* F2re = mats + M_F2RE;
  const float* F2im = mats + M_F2IM;
  const int total = BB * CC * NN * NFQ;
  for (int idx = blockIdx.x * blockDim.x + threadIdx.x; idx < total;
       idx += gridDim.x * blockDim.x) {
    int img = idx / (NN * NFQ);
    int rem = idx % (NN * NFQ);
    int k = rem / NFQ, f = rem % NFQ;
    int c = img % CC, b = img / CC;
    const float* yr = yre + (size_t)img * HH * NFQ + f;
    const float* yi = yim + (size_t)img * HH * NFQ + f;
    float re = 0.0f, im = 0.0f;
    for (int r = 0; r < HH; ++r) {
      float cr = F2re[k * HH + r], ci = F2im[k * HH + r];
      float vr = yr[(size_t)r * NFQ], vi = yi[(size_t)r * NFQ];
      re += cr * vr - ci * vi;
      im += cr * vi + ci * vr;
    }
    size_t o = (size_t)rem * CC * BB + (size_t)c * BB + b;
    xfre[o] = re;
    xfim[o] = im;
  }
}

// ---------------------------------------------------------------------------
__global__ void ocn_build_V(const float* __restrict__ w, int qbase, float* __restrict__ scr) {
  int q = qbase + blockIdx.x;
  int k = q / NFQ, f = q % NFQ;
  float* Vre = scr + (size_t)blockIdx.x * PERQ + PQ_VRE;
  float* Vim = scr + (size_t)blockIdx.x * PERQ + PQ_VIM;
  int o = threadIdx.x;
  float cs[9], sn[9];
#pragma unroll
  for (int a = 0; a < 3; ++a)
#pragma unroll
    for (int b = 0; b < 3; ++b) {
      float th = TWO_PI * (float)(a * k + b * f) / 34.0f;
      cs[a * 3 + b] = cosf(th);
      sn[a * 3 + b] = sinf(th);
    }
  const float* wrow = w + (size_t)o * 255 * 9;
  for (int i = 0; i < 255; ++i) {
    float re = 0.0f, im = 0.0f;
#pragma unroll
    for (int t = 0; t < 9; ++t) {
      float wv = wrow[i * 9 + t];
      re += wv * cs[t];
      im -= wv * sn[t];
    }
    Vre[o * 256 + i] = re * (1.0f / 34.0f);
    Vim[o * 256 + i] = im * (1.0f / 34.0f);
  }
  Vre[o * 256 + 255] = 0.0f;
  Vim[o * 256 + 255] = 0.0f;
}

// ---------------------------------------------------------------------------
// A = V^H V (256x256 padded Hermitian) via fp32 WMMA.  grid=(CHUNK,16)
__global__ void ocn_build_A(float* __restrict__ scr) {
  float* base = scr + (size_t)blockIdx.x * PERQ;
  const float* Vre = base + PQ_VRE;
  const float* Vim = base + PQ_VIM;
  float* Are = base + PQ_ARE;
  float* Aim = base + PQ_AIM;

  int tid = threadIdx.x, wave = tid >> 5, lane = tid & 31;
  int half = lane >> 4, mrow = lane & 15, koff = half * 2, ncol = lane & 15;
  int rowBase = blockIdx.y * 16;
  int colBase = wave * 32;
  int rm = rowBase + mrow;
  int c0 = colBase + ncol, c1 = colBase + 16 + ncol;

  v8f are0 = {}, are1 = {}, aim0 = {}, aim1 = {};
  for (int kb = 0; kb < 256; kb += 4) {
    int kk = kb + koff;
    v2f ar;  ar.x = Vre[kk * 256 + rm];  ar.y = Vre[(kk + 1) * 256 + rm];
    v2f ai;  ai.x = Vim[kk * 256 + rm];  ai.y = Vim[(kk + 1) * 256 + rm];
    v2f ain; ain.x = -ai.x;              ain.y = -ai.y;
    v2f br0; br0.x = Vre[kk * 256 + c0]; br0.y = Vre[(kk + 1) * 256 + c0];
    v2f bi0; bi0.x = Vim[kk * 256 + c0]; bi0.y = Vim[(kk + 1) * 256 + c0];
    v2f br1; br1.x = Vre[kk * 256 + c1]; br1.y = Vre[(kk + 1) * 256 + c1];
    v2f bi1; bi1.x = Vim[kk * 256 + c1]; bi1.y = Vim[(kk + 1) * 256 + c1];
    are0 = wmma4(ar, br0, are0);  are0 = wmma4(ai, bi0, are0);
    aim0 = wmma4(ar, bi0, aim0);  aim0 = wmma4(ain, br0, aim0);
    are1 = wmma4(ar, br1, are1);  are1 = wmma4(ai, bi1, are1);
    aim1 = wmma4(ar, bi1, aim1);  aim1 = wmma4(ain, br1, aim1);
  }
#pragma unroll
  for (int i = 0; i < 8; ++i) {
    int r = rowBase + i + half * 8;
    Are[r * 256 + c0] = are0[i];
    Aim[r * 256 + c0] = aim0[i];
    Are[r * 256 + c1] = are1[i];
    Aim[r * 256 + c1] = aim1[i];
  }
}

// ---------------------------------------------------------------------------
__global__ void ocn_add_eps(float* __restrict__ scr) {
  float* base = scr + (size_t)blockIdx.x * PERQ;
  float* Are = base + PQ_ARE;
  float* Aim = base + PQ_AIM;
  __shared__ float red[256];
  int t = threadIdx.x;
  red[t] = (t < 255) ? Are[t * 257] : 0.0f;
  __syncthreads();
  for (int s = 128; s > 0; s >>= 1) {
    if (t < s) red[t] += red[t + s];
    __syncthreads();
  }
  float eps = red[0] * (1.0f / 255.0f) * 1e-5f;
  if (t < 255) Are[t * 257] += eps;
  if (t == 255) Are[255 * 257] = 1.0f;
  Aim[t * 257] = 0.0f;
}

// ---------------------------------------------------------------------------
__global__ void ocn_cholesky(float* __restrict__ scr) {
  float* base = scr + (size_t)blockIdx.x * PERQ;
  float* Are = base + PQ_ARE;
  float* Aim = base + PQ_AIM;
  int t = threadIdx.x;
  for (int k = 0; k < 256; ++k) {
    if (t == 0) Are[k * 257] = sqrtf(fmaxf(Are[k * 257], 1e-30f));
    __threadfence();
    __syncthreads();
    float dinv = 1.0f / Are[k * 257];
    if (t > k) {
      Are[t * 256 + k] *= dinv;
      Aim[t * 256 + k] *= dinv;
    }
    __threadfence();
    __syncthreads();
    if (t > k) {
      // prefetch the column we will need on the next outer iteration
      __builtin_prefetch(&Are[t * 256 + k + 1], 0, 1);
      __builtin_prefetch(&Aim[t * 256 + k + 1], 0, 1);
      float ljr = Are[t * 256 + k], lji = Aim[t * 256 + k];
      for (int i = k + 1; i <= t; ++i) {
        float lir = Are[i * 256 + k], lii = Aim[i * 256 + k];
        float pr = ljr * lir + lji * lii;
        float pi = lji * lir - ljr * lii;
        Are[t * 256 + i] -= pr;
        Aim[t * 256 + i] -= pi;
      }
    }
    __threadfence();
    __syncthreads();
  }
}

// ---------------------------------------------------------------------------
// T = V^H X (256x64), WMMA; X_q staged to LDS via the Tensor Data Mover.
// grid=(CHUNK,2), dynamic LDS = 128KB.
__global__ void ocn_compute_T(float* __restrict__ scr, const float* __restrict__ xfre,
                              const float* __restrict__ xfim, int qbase) {
  extern __shared__ float xs[];  // [2][256][64]
  float* base = scr + (size_t)blockIdx.x * PERQ;
  const float* Vre = base + PQ_VRE;
  const float* Vim = base + PQ_VIM;
  float* Tre = base + PQ_TRE;
  float* Tim = base + PQ_TIM;
  size_t qcb = (size_t)(qbase + blockIdx.x) * CC * BB;

  int tid = threadIdx.x, wave = tid >> 5, lane = tid & 31;
  int half = lane >> 4, mrow = lane & 15, koff = half * 2, ncol = lane & 15;

#ifdef OCN_TDM
  // each of the 8 waves DMAs its 32-row slice (2048 f32) of both planes
  unsigned lb = __builtin_amdgcn_groupstaticsize();
  ocn_tdm_load_1d(xfre + qcb + (size_t)wave * 2048, lb + (unsigned)wave * 8192u, 2048);
  ocn_tdm_load_1d(xfim + qcb + (size_t)wave * 2048, lb + 65536u + (unsigned)wave * 8192u, 2048);
  __builtin_amdgcn_s_wait_tensorcnt(0);
  __syncthreads();
#else
  for (int i = tid; i < 16384; i += 256) {
    xs[i] = xfre[qcb + i];
    xs[16384 + i] = xfim[qcb + i];
  }
  __syncthreads();
#endif

  int rowBase = blockIdx.y * 128 + wave * 16;
  int rm = rowBase + mrow;

  v8f tre[4] = {{}, {}, {}, {}};
  v8f tim[4] = {{}, {}, {}, {}};
  for (int kb = 0; kb < 256; kb += 4) {
    int kk = kb + koff;
    v2f ar;  ar.x = Vre[kk * 256 + rm];  ar.y = Vre[(kk + 1) * 256 + rm];
    v2f ai;  ai.x = Vim[kk * 256 + rm];  ai.y = Vim[(kk + 1) * 256 + rm];
    v2f ain; ain.x = -ai.x;              ain.y = -ai.y;
#pragma unroll
    for (int ct = 0; ct < 4; ++ct) {
      int col = ct * 16 + ncol;
      v2f bxr; bxr.x = xs[kk * BB + col];         bxr.y = xs[(kk + 1) * BB + col];
      v2f bxi; bxi.x = xs[16384 + kk * BB + col]; bxi.y = xs[16384 + (kk + 1) * BB + col];
      tre[ct] = wmma4(ar, bxr, tre[ct]);  tre[ct] = wmma4(ai, bxi, tre[ct]);
      tim[ct] = wmma4(ar, bxi, tim[ct]);  tim[ct] = wmma4(ain, bxr, tim[ct]);
    }
  }
#pragma unroll
  for (int ct = 0; ct < 4; ++ct) {
    int col = ct * 16 + ncol;
#pragma unroll
    for (int i = 0; i < 8; ++i) {
      int r = rowBase + i + half * 8;
      Tre[r * BB + col] = tre[ct][i];
      Tim[r * BB + col] = tim[ct][i];
    }
  }
}

// ---------------------------------------------------------------------------
__global__ void ocn_solve(float* __restrict__ scr) {
  float* base = scr + (size_t)blockIdx.x * PERQ;
  const float* Are = base + PQ_ARE;
  const float* Aim = base + PQ_AIM;
  const float* Tre = base + PQ_TRE;
  const float* Tim = base + PQ_TIM;
  float* Sre = base + PQ_SRE;
  float* Sim = base + PQ_SIM;
  int c = threadIdx.x;  // 0..63
  for (int r = 0; r < 256; ++r) {
    __builtin_prefetch(&Are[(r + 1) * 256], 0, 1);
    __builtin_prefetch(&Aim[(r + 1) * 256], 0, 1);
    float ar = Tre[r * BB + c], ai = Tim[r * BB + c];
    for (int j = 0; j < r; ++j) {
      float lr = Are[r * 256 + j], li = Aim[r * 256 + j];
      float yr = Sre[j * BB + c], yi = Sim[j * BB + c];
      ar -= lr * yr - li * yi;
      ai -= lr * yi + li * yr;
    }
    float dinv = 1.0f / Are[r * 257];
    Sre[r * BB + c] = ar * dinv;
    Sim[r * BB + c] = ai * dinv;
  }
  for (int r = 255; r >= 0; --r) {
    float ar = Sre[r * BB + c], ai = Sim[r * BB + c];
    for (int j = r + 1; j < 256; ++j) {
      float lr = Are[j * 256 + r], li = -Aim[j * 256 + r];
      float sr = Sre[j * BB + c], si = Sim[j * BB + c];
      ar -= lr * sr - li * si;
      ai -= lr * si + li * sr;
    }
    float dinv = 1.0f / Are[r * 257];
    Sre[r * BB + c] = ar * dinv;
    Sim[r * BB + c] = ai * dinv;
  }
}

// ---------------------------------------------------------------------------
// Z = X - 2 V S (256x64), WMMA; S staged to LDS via TDM.  grid=(CHUNK,2)
__global__ void ocn_compute_Z(float* __restrict__ scr, const float* __restrict__ xfre,
                              const float* __restrict__ xfim, float* __restrict__ zfre,
                              float* __restrict__ zfim, int qbase) {
  extern __shared__ float ss[];  // [2][256][64]
  float* base = scr + (size_t)blockIdx.x * PERQ;
  const float* Vre = base + PQ_VRE;
  const float* Vim = base + PQ_VIM;
  const float* Sre = base + PQ_SRE;
  const float* Sim = base + PQ_SIM;
  size_t qcb = (size_t)(qbase + blockIdx.x) * CC * BB;

  int tid = threadIdx.x, wave = tid >> 5, lane = tid & 31;
  int half = lane >> 4, mrow = lane & 15, koff = half * 2, ncol = lane & 15;

#ifdef OCN_TDM
  unsigned lb = __builtin_amdgcn_groupstaticsize();
  ocn_tdm_load_1d(Sre + (size_t)wave * 2048, lb + (unsigned)wave * 8192u, 2048);
  ocn_tdm_load_1d(Sim + (size_t)wave * 2048, lb + 65536u + (unsigned)wave * 8192u, 2048);
  __builtin_amdgcn_s_wait_tensorcnt(0);
  __syncthreads();
#else
  for (int i = tid; i < 16384; i += 256) {
    ss[i] = Sre[i];
    ss[16384 + i] = Sim[i];
  }
  __syncthreads();
#endif

  int rowBase = blockIdx.y * 128 + wave * 16;
  int rm = rowBase + mrow;

  v8f zre[4] = {{}, {}, {}, {}};
  v8f zim[4] = {{}, {}, {}, {}};
  for (int kb = 0; kb < 256; kb += 4) {
    int kk = kb + koff;
    v2f ar;  ar.x = Vre[rm * 256 + kk];  ar.y = Vre[rm * 256 + kk + 1];
    v2f ai;  ai.x = Vim[rm * 256 + kk];  ai.y = Vim[rm * 256 + kk + 1];
    v2f ain; ain.x = -ai.x;              ain.y = -ai.y;
#pragma unroll
    for (int ct = 0; ct < 4; ++ct) {
      int col = ct * 16 + ncol;
      v2f bsr; bsr.x = ss[kk * BB + col];         bsr.y = ss[(kk + 1) * BB + col];
      v2f bsi; bsi.x = ss[16384 + kk * BB + col]; bsi.y = ss[16384 + (kk + 1) * BB + col];
      zre[ct] = wmma4(ar, bsr, zre[ct]);  zre[ct] = wmma4(ain, bsi, zre[ct]);
      zim[ct] = wmma4(ar, bsi, zim[ct]);  zim[ct] = wmma4(ai, bsr, zim[ct]);
    }
  }
#pragma unroll
  for (int ct = 0; ct < 4; ++ct) {
    int col = ct * 16 + ncol;
#pragma unroll
    for (int i = 0; i < 8; ++i) {
      int r = rowBase + i + half * 8;
      size_t o = qcb + (size_t)r * BB + col;
      zfre[o] = xfre[o] - 2.0f * zre[ct][i];
      zfim[o] = xfim[o] - 2.0f * zim[ct][i];
    }
  }
}

// ---------------------------------------------------------------------------
__global__ void ocn_inv1(const float* __restrict__ zfre, const float* __restrict__ zfim,
                         const float* __restrict__ mats,
                         float* __restrict__ yre, float* __restrict__ yim) {
  const float* G1re = mats + M_G1RE;
  const float* G1im = mats + M_G1IM;
  const int total = BB * CC * HH * NFQ;
  for (int idx = blockIdx.x * blockDim.x + threadIdx.x; idx < total;
       idx += gridDim.x * blockDim.x) {
    int img = idx / (HH * NFQ);
    int rem = idx % (HH * NFQ);
    int r = rem / NFQ, f = rem % NFQ;
    int c = img % CC, b = img / CC;
    float re = 0.0f, im = 0.0f;
    for (int k = 0; k < NN; ++k) {
      float gr = G1re[r * NN + k], gi = G1im[r * NN + k];
      size_t o = (size_t)(k * NFQ + f) * CC * BB + (size_t)c * BB + b;
      float zr = zfre[o], zi = zfim[o];
      re += gr * zr - gi * zi;
      im += gr * zi + gi * zr;
    }
    yre[(size_t)img * HH * NFQ + rem] = re;
    yim[(size_t)img * HH * NFQ + rem] = im;
  }
}

// ---------------------------------------------------------------------------
__global__ void ocn_inv2(const float* __restrict__ yre, const float* __restrict__ yim,
                         const float* __restrict__ mats, const float* __restrict__ bias,
                         float* __restrict__ out) {
  const float* C2re = mats + M_C2RE;
  const float* C2im = mats + M_C2IM;
  const int total = BB * CC * HH * HH;
  for (int idx = blockIdx.x * blockDim.x + threadIdx.x; idx < total;
       idx += gridDim.x * blockDim.x) {
    int img = idx / (HH * HH);
    int rem = idx % (HH * HH);
    int r = rem / HH, cc2 = rem % HH;
    int c = img % CC;
    const float* y2r = yre + (size_t)img * HH * NFQ + (size_t)r * NFQ;
    const float* y2i = yim + (size_t)img * HH * NFQ + (size_t)r * NFQ;
    float acc = bias[c];
    for (int f = 0; f < NFQ; ++f)
      acc += y2r[f] * C2re[f * HH + cc2] + y2i[f] * C2im[f * HH + cc2];
    out[idx] = acc;
  }
}

// ---------------------------------------------------------------------------
extern "C" void kernel_launch(void* const* d_in, const int* in_sizes, int n_in,
                              void* d_out, int out_size, void* d_ws, size_t ws_size,
                              hipStream_t stream) {
  (void)in_sizes; (void)n_in; (void)out_size; (void)ws_size;
  const float* x    = (const float*)d_in[0];
  const float* w    = (const float*)d_in[1];
  const float* bias = (const float*)d_in[2];
  float* ws   = (float*)d_ws;
  float* mats = ws + W_MATS;
  float* xfre = ws + W_XFRE;
  float* xfim = ws + W_XFIM;
  float* zfre = ws + W_ZFRE;
  float* zfim = ws + W_ZFIM;
  float* yre  = ws + W_YRE;
  float* yim  = ws + W_YIM;
  float* scr  = ws + W_SCR;

  ocn_init_dft<<<1, 256, 0, stream>>>(mats);
  ocn_fwd1<<<4096, 256, 0, stream>>>(x, mats, yre, yim);
  ocn_fwd2<<<8192, 256, 0, stream>>>(yre, yim, mats, xfre, xfim);

  for (int qb = 0; qb < QQ; qb += CHUNK) {
    ocn_build_V<<<CHUNK, 256, 0, stream>>>(w, qb, scr);
    ocn_build_A<<<dim3(CHUNK, 16), 256, 0, stream>>>(scr);
    ocn_add_eps<<<CHUNK, 256, 0, stream>>>(scr);
    ocn_cholesky<<<CHUNK, 256, 0, stream>>>(scr);
    ocn_compute_T<<<dim3(CHUNK, 2), 256, 131072, stream>>>(scr, xfre, xfim, qb);
    ocn_solve<<<CHUNK, 64, 0, stream>>>(scr);
    ocn_compute_Z<<<dim3(CHUNK, 2), 256, 131072, stream>>>(scr, xfre, xfim, zfre, zfim, qb);
  }

  ocn_inv1<<<8192, 256, 0, stream>>>(zfre, zfim, mats, yre, yim);
  ocn_inv2<<<16384, 256, 0, stream>>>(yre, yim, mats, bias, (float*)d_out);
}